// Messages_82892868812884
// MI455X (gfx1250) — compile-verified
//
#include <hip/hip_runtime.h>
#include <math.h>

#define NNODES 50000
#define NEDGES 500000
#define CHAN   64
#define R0f    2.0f

#define WAVES  8
#define BLOCK  (WAVES * 32)
#define NTILES (NEDGES / 16)
#define GRID   512

typedef float v2f __attribute__((ext_vector_type(2)));
typedef float v8f __attribute__((ext_vector_type(8)));

// ---- LDS layout (float offsets) ----
// Weight matrices stored TRANSPOSED: wT[n][k] so B-operand (k,k+1) at fixed n is contiguous.
#define OFF_WSRC  0                     // 64x64 -> 4096
#define OFF_WDST  4096                  // 4096
#define OFF_WS    8192                  // 4096
#define OFF_WV    12288                 // 4096
#define OFF_WGATE 16384                 // 256x64 -> 16384
#define OFF_WENC  32768                 // 64x8 -> 512
#define OFF_BIAS0 33280                 // 64 (b_src+b_dst+b_enc)
#define OFF_BGATE 33344                 // 256
#define OFF_STAGE 33600                 // WAVES * 1024 (16x64 f32 per wave)
#define OFF_RH    (OFF_STAGE + WAVES * 1024)   // WAVES * 64  (16 x 4 floats)
#define OFF_IDX   (OFF_RH + WAVES * 64)        // WAVES * 32  (16 src + 16 dst ints)
#define SMEM_FLOATS (OFF_IDX + WAVES * 32)     // 42560 floats = 170240 B (< 320KB WGP LDS)

__device__ __forceinline__ void lds_fence() {
  // LDS is processed in-order per wave; this both drains DScnt and stops the
  // compiler from reordering LDS traffic around the staging-buffer reuse.
  asm volatile("s_wait_dscnt 0x0" ::: "memory");
}

__device__ __forceinline__ v8f wmma4(v2f a, v2f b, v8f c) {
  // V_WMMA_F32_16X16X4_F32 : D(16x16,f32) = A(16x4,f32) * B(4x16,f32) + C
  return __builtin_amdgcn_wmma_f32_16x16x4_f32(false, a, false, b, (short)0, c,
                                               false, false);
}

__global__ void zero_f32(float* __restrict__ p, int n) {
  int i = blockIdx.x * blockDim.x + threadIdx.x;
  if (i < n) p[i] = 0.0f;
}

__global__ void __launch_bounds__(BLOCK, 1)
messages_kernel(const int* __restrict__ src, const int* __restrict__ dst,
                const float* __restrict__ r_ij,
                const float* __restrict__ z0g, const float* __restrict__ z1g,
                const float* __restrict__ emb,
                const float* __restrict__ Wenc, const float* __restrict__ benc,
                const float* __restrict__ Wsrc, const float* __restrict__ bsrc,
                const float* __restrict__ Wdst, const float* __restrict__ bdst,
                const float* __restrict__ Wgate, const float* __restrict__ bgate,
                const float* __restrict__ Ws, const float* __restrict__ Wv,
                float* __restrict__ out0, float* __restrict__ out1) {
  extern __shared__ float smem[];

  // -------- cooperative weight staging (transposed) --------
  for (int i = threadIdx.x; i < 64 * 64; i += BLOCK) {
    int k = i >> 6, n = i & 63;
    smem[OFF_WSRC + n * 64 + k] = Wsrc[i];
    smem[OFF_WDST + n * 64 + k] = Wdst[i];
    smem[OFF_WS   + n * 64 + k] = Ws[i];
    smem[OFF_WV   + n * 64 + k] = Wv[i];
  }
  for (int i = threadIdx.x; i < 64 * 256; i += BLOCK) {
    int k = i >> 8, n = i & 255;
    smem[OFF_WGATE + n * 64 + k] = Wgate[i];
  }
  for (int i = threadIdx.x; i < 8 * 64; i += BLOCK) {
    int k = i >> 6, n = i & 63;
    smem[OFF_WENC + n * 8 + k] = Wenc[i];
  }
  if (threadIdx.x < 64)
    smem[OFF_BIAS0 + threadIdx.x] =
        bsrc[threadIdx.x] + bdst[threadIdx.x] + benc[threadIdx.x];
  if (threadIdx.x < 256) smem[OFF_BGATE + threadIdx.x] = bgate[threadIdx.x];
  __syncthreads();

  const int lane = threadIdx.x & 31;
  const int wave = threadIdx.x >> 5;
  const int half = lane >> 4;   // k-half selector for A/B operands
  const int nl   = lane & 15;   // edge-row (A side) / channel-lane (C side)

  float* stage = smem + OFF_STAGE + wave * 1024;           // 16x64 row-major
  float* rhS   = smem + OFF_RH + wave * 64;                // 16 x 4 (rhat xyz)
  int*   idxS  = ((int*)(smem + OFF_IDX)) + wave * 32;     // 16 src, 16 dst

  for (int tile = blockIdx.x * WAVES + wave; tile < NTILES;
       tile += gridDim.x * WAVES) {
    const int e  = tile * 16 + nl;   // this lane's edge row (dup across halves)
    const int sN = src[e];
    const int dN = dst[e];
    if (half == 0) { idxS[nl] = sN; idxS[16 + nl] = dN; }

    // -------- geometry: d, radial basis, r_hat --------
    float rx = r_ij[e * 3 + 0], ry = r_ij[e * 3 + 1], rz = r_ij[e * 3 + 2];
    float dd  = sqrtf(rx * rx + ry * ry + rz * rz);
    float sx = rx * 3.5f, sy = ry * 3.5f, sz = rz * 3.5f;   // 7/R0
    float inv = rsqrtf(1.0f + sx * sx + sy * sy + sz * sz);
    if (half == 0) {
      rhS[nl * 4 + 0] = sx * inv;
      rhS[nl * 4 + 1] = sy * inv;
      rhS[nl * 4 + 2] = sz * inv;
    }
    float rb[4];   // this lane's 4 radial-basis values in A layout order
#pragma unroll
    for (int u = 0; u < 4; ++u) {
      int k = (u >> 1) * 4 + 2 * half + (u & 1);
      float x = (dd - (float)k * (R0f / 7.0f)) * 4.0f;   // sigma = R0/8 = 0.25
      rb[u] = expf(-x * x);
    }

    // -------- dist_emb = emb[s]@Wsrc + emb[d]@Wdst + rb@Wenc + bias --------
    v8f acc[4];
#pragma unroll
    for (int j = 0; j < 4; ++j) acc[j] = (v8f)0.0f;

    const float* es = emb + (size_t)sN * 64;
    const float* ed = emb + (size_t)dN * 64;
#pragma unroll
    for (int t = 0; t < 16; ++t) {
      int k0 = 4 * t + 2 * half;
      v2f a = *(const v2f*)(es + k0);
#pragma unroll
      for (int j = 0; j < 4; ++j) {
        v2f b = *(const v2f*)(smem + OFF_WSRC + (16 * j + nl) * 64 + k0);
        acc[j] = wmma4(a, b, acc[j]);
      }
    }
#pragma unroll
    for (int t = 0; t < 16; ++t) {
      int k0 = 4 * t + 2 * half;
      v2f a = *(const v2f*)(ed + k0);
#pragma unroll
      for (int j = 0; j < 4; ++j) {
        v2f b = *(const v2f*)(smem + OFF_WDST + (16 * j + nl) * 64 + k0);
        acc[j] = wmma4(a, b, acc[j]);
      }
    }
#pragma unroll
    for (int t = 0; t < 2; ++t) {
      int k0 = 4 * t + 2 * half;
      v2f a;
      a[0] = rb[2 * t];
      a[1] = rb[2 * t + 1];
#pragma unroll
      for (int j = 0; j < 4; ++j) {
        v2f b = *(const v2f*)(smem + OFF_WENC + (16 * j + nl) * 8 + k0);
        acc[j] = wmma4(a, b, acc[j]);
      }
    }
#pragma unroll
    for (int j = 0; j < 4; ++j) {
      float bb = smem[OFF_BIAS0 + 16 * j + nl];
#pragma unroll
      for (int r = 0; r < 8; ++r) acc[j][r] += bb;
    }

    // -------- stage dist_emb (C layout -> row-major 16x64) --------
    lds_fence();   // previous iter's reads of stage are done before overwrite
#pragma unroll
    for (int j = 0; j < 4; ++j)
#pragma unroll
      for (int r = 0; r < 8; ++r)
        stage[(r + 8 * half) * 64 + 16 * j + nl] = acc[j][r];
    lds_fence();

    // -------- gather per-row ids, rhat, z0, z1 (C layout) --------
    int sn[8], dn[8];
#pragma unroll
    for (int r = 0; r < 8; ++r) {
      sn[r] = idxS[r + 8 * half];
      dn[r] = idxS[16 + r + 8 * half];
    }
    float rh[3][8];
#pragma unroll
    for (int r = 0; r < 8; ++r) {
      rh[0][r] = rhS[(r + 8 * half) * 4 + 0];
      rh[1][r] = rhS[(r + 8 * half) * 4 + 1];
      rh[2][r] = rhS[(r + 8 * half) * 4 + 2];
    }
    float z0r[4][8];
#pragma unroll
    for (int j = 0; j < 4; ++j)
#pragma unroll
      for (int r = 0; r < 8; ++r)
        z0r[j][r] = z0g[(size_t)dn[r] * 64 + 16 * j + nl];
    float z1r[3][4][8];
#pragma unroll
    for (int i = 0; i < 3; ++i)
#pragma unroll
      for (int j = 0; j < 4; ++j)
#pragma unroll
        for (int r = 0; r < 8; ++r)
          z1r[i][j][r] = z1g[(size_t)dn[r] * 192 + i * 64 + 16 * j + nl];

    // -------- gate = dist_emb @ Wgate + bgate, 4 chunks of 64 --------
    v8f g[4][4];
#pragma unroll
    for (int c = 0; c < 4; ++c) {
#pragma unroll
      for (int j = 0; j < 4; ++j) g[c][j] = (v8f)0.0f;
#pragma unroll
      for (int t = 0; t < 16; ++t) {
        int k0 = 4 * t + 2 * half;
        v2f a = *(const v2f*)(stage + nl * 64 + k0);
#pragma unroll
        for (int j = 0; j < 4; ++j) {
          v2f b = *(const v2f*)(smem + OFF_WGATE +
                                (64 * c + 16 * j + nl) * 64 + k0);
          g[c][j] = wmma4(a, b, g[c][j]);
        }
      }
#pragma unroll
      for (int j = 0; j < 4; ++j) {
        float bb = smem[OFF_BGATE + 64 * c + 16 * j + nl];
#pragma unroll
        for (int r = 0; r < 8; ++r) g[c][j][r] += bb;
      }
    }

    // -------- s_msg = g0*z0 + g1*(rhat . z1)   (overwrite g[0]) --------
#pragma unroll
    for (int j = 0; j < 4; ++j)
#pragma unroll
      for (int r = 0; r < 8; ++r) {
        float dotv = rh[0][r] * z1r[0][j][r] + rh[1][r] * z1r[1][j][r] +
                     rh[2][r] * z1r[2][j][r];
        g[0][j][r] = g[0][j][r] * z0r[j][r] + g[1][j][r] * dotv;
      }

    // -------- psi0 = s_msg @ Ws ; scatter to out0[src] --------
    lds_fence();
#pragma unroll
    for (int j = 0; j < 4; ++j)
#pragma unroll
      for (int r = 0; r < 8; ++r)
        stage[(r + 8 * half) * 64 + 16 * j + nl] = g[0][j][r];
    lds_fence();
    v8f p[4];
#pragma unroll
    for (int j = 0; j < 4; ++j) p[j] = (v8f)0.0f;
#pragma unroll
    for (int t = 0; t < 16; ++t) {
      int k0 = 4 * t + 2 * half;
      v2f a = *(const v2f*)(stage + nl * 64 + k0);
#pragma unroll
      for (int j = 0; j < 4; ++j) {
        v2f b = *(const v2f*)(smem + OFF_WS + (16 * j + nl) * 64 + k0);
        p[j] = wmma4(a, b, p[j]);
      }
    }
#pragma unroll
    for (int j = 0; j < 4; ++j)
#pragma unroll
      for (int r = 0; r < 8; ++r)
        atomicAdd(&out0[(size_t)sn[r] * 64 + 16 * j + nl], p[j][r]);

    // -------- psi1[i] = (g2*z1[i] + g3*rhat_i*z0) @ Wv ; scatter --------
#pragma unroll
    for (int i = 0; i < 3; ++i) {
      lds_fence();
#pragma unroll
      for (int j = 0; j < 4; ++j)
#pragma unroll
        for (int r = 0; r < 8; ++r)
          stage[(r + 8 * half) * 64 + 16 * j + nl] =
              g[2][j][r] * z1r[i][j][r] + g[3][j][r] * rh[i][r] * z0r[j][r];
      lds_fence();
#pragma unroll
      for (int j = 0; j < 4; ++j) p[j] = (v8f)0.0f;
#pragma unroll
      for (int t = 0; t < 16; ++t) {
        int k0 = 4 * t + 2 * half;
        v2f a = *(const v2f*)(stage + nl * 64 + k0);
#pragma unroll
        for (int j = 0; j < 4; ++j) {
          v2f b = *(const v2f*)(smem + OFF_WV + (16 * j + nl) * 64 + k0);
          p[j] = wmma4(a, b, p[j]);
        }
      }
#pragma unroll
      for (int j = 0; j < 4; ++j)
#pragma unroll
        for (int r = 0; r < 8; ++r)
          atomicAdd(&out1[(size_t)sn[r] * 192 + i * 64 + 16 * j + nl],
                    p[j][r]);
    }
  }
}

extern "C" void kernel_launch(void* const* d_in, const int* in_sizes, int n_in,
                              void* d_out, int out_size, void* d_ws,
                              size_t ws_size, hipStream_t stream) {
  const int*   src   = (const int*)d_in[0];
  const int*   dst   = (const int*)d_in[1];
  const float* r_ij  = (const float*)d_in[2];
  const float* z0    = (const float*)d_in[3];
  const float* z1    = (const float*)d_in[4];
  const float* emb   = (const float*)d_in[5];
  const float* Wenc  = (const float*)d_in[6];
  const float* benc  = (const float*)d_in[7];
  const float* Wsrc  = (const float*)d_in[8];
  const float* bsrc  = (const float*)d_in[9];
  const float* Wdst  = (const float*)d_in[10];
  const float* bdst  = (const float*)d_in[11];
  const float* Wgate = (const float*)d_in[12];
  const float* bgate = (const float*)d_in[13];
  const float* Ws    = (const float*)d_in[14];
  const float* Wv    = (const float*)d_in[15];
  float* out = (float*)d_out;

  // harness poisons d_out; atomics need zeros
  zero_f32<<<(out_size + 255) / 256, 256, 0, stream>>>(out, out_size);

  size_t smem_bytes = (size_t)SMEM_FLOATS * sizeof(float);
  messages_kernel<<<GRID, BLOCK, smem_bytes, stream>>>(
      src, dst, r_ij, z0, z1, emb, Wenc, benc, Wsrc, bsrc, Wdst, bdst, Wgate,
      bgate, Ws, Wv, out, out + (size_t)NNODES * CHAN);
}